// CustomModel_4827543241131
// MI455X (gfx1250) — compile-verified
//
#include <hip/hip_runtime.h>
#include <hip/hip_bf16.h>
#include <math.h>

// ---- problem constants (from reference) ----
#define BS    16
#define CAP   64
#define VOCAB 50265
#define TOPK  5
#define SEQ   256
#define AT    32
#define DIM   768
#define END_ID 2

typedef __attribute__((ext_vector_type(2))) float v2f;
typedef __attribute__((ext_vector_type(8))) float v8f;

// =====================================================================
// Kernel 1: per-(b,c) row: top-5 over VOCAB logits, softmax, and
// sce[b,c,:] = sum_k prob[k] * embed_table[ids[k], :]
// One 256-thread workgroup per row; coalesced streaming scan (the
// bandwidth-dominant 206 MB read), register top-5 + LDS tree merge.
// =====================================================================
__device__ __forceinline__ bool better(float a, int ia, float b, int ib) {
    return (a > b) || (a == b && ia < ib);
}

__global__ void topk_sce_kernel(const float* __restrict__ cap_out,
                                const float* __restrict__ embed,
                                float* __restrict__ sce)
{
    const int row = blockIdx.x;                 // b*CAP + c
    const float* __restrict__ logits = cap_out + (size_t)row * VOCAB;
    const int tid = threadIdx.x;                // 0..255

    float v[TOPK];
    int   id[TOPK];
#pragma unroll
    for (int j = 0; j < TOPK; ++j) { v[j] = -INFINITY; id[j] = 0x7fffffff; }

    for (int i = tid; i < VOCAB; i += 256) {
        if (i + 1024 < VOCAB) __builtin_prefetch(&logits[i + 1024], 0, 0);
        float x = logits[i];
        if (better(x, i, v[TOPK - 1], id[TOPK - 1])) {
            int j = TOPK - 1;
            while (j > 0 && better(x, i, v[j - 1], id[j - 1])) {
                v[j] = v[j - 1]; id[j] = id[j - 1]; --j;
            }
            v[j] = x; id[j] = i;
        }
    }

    __shared__ float sv[256][TOPK];
    __shared__ int   si[256][TOPK];
#pragma unroll
    for (int j = 0; j < TOPK; ++j) { sv[tid][j] = v[j]; si[tid][j] = id[j]; }
    __syncthreads();

    // tree merge of sorted-descending 5-lists
    for (int stride = 128; stride >= 1; stride >>= 1) {
        if (tid < stride) {
            float va[TOPK], vb[TOPK], rv[TOPK];
            int   ia[TOPK], ib[TOPK], ri[TOPK];
#pragma unroll
            for (int j = 0; j < TOPK; ++j) {
                va[j] = sv[tid][j];          ia[j] = si[tid][j];
                vb[j] = sv[tid + stride][j]; ib[j] = si[tid + stride][j];
            }
            int pa = 0, pb = 0;
#pragma unroll
            for (int o = 0; o < TOPK; ++o) {
                bool takeA = (pb >= TOPK) ||
                             (pa < TOPK && better(va[pa], ia[pa], vb[pb], ib[pb]));
                if (takeA) { rv[o] = va[pa]; ri[o] = ia[pa]; ++pa; }
                else       { rv[o] = vb[pb]; ri[o] = ib[pb]; ++pb; }
            }
#pragma unroll
            for (int j = 0; j < TOPK; ++j) { sv[tid][j] = rv[j]; si[tid][j] = ri[j]; }
        }
        __syncthreads();
    }

    __shared__ float prob[TOPK];
    __shared__ int   topid[TOPK];
    if (tid == 0) {
        float mx = sv[0][0];                 // sorted descending -> max first
        float sum = 0.f, e[TOPK];
#pragma unroll
        for (int k = 0; k < TOPK; ++k) { e[k] = __expf(sv[0][k] - mx); sum += e[k]; }
        float inv = 1.0f / sum;
#pragma unroll
        for (int k = 0; k < TOPK; ++k) { prob[k] = e[k] * inv; topid[k] = si[0][k]; }
    }
    __syncthreads();

    float* __restrict__ out = sce + (size_t)row * DIM;
    for (int d = tid; d < DIM; d += 256) {
        float acc = 0.f;
#pragma unroll
        for (int k = 0; k < TOPK; ++k)
            acc += prob[k] * embed[(size_t)topid[k] * DIM + d];
        out[d] = acc;
    }
}

// =====================================================================
// Kernel 2: attention via V_WMMA_F32_16X16X4_F32 (f32-exact vs reference).
// One wave32 per (batch, 16-row M-tile of AT).
//   GEMM1: scores[16,64] = Q[16,768] x SCE[64,768]^T   (4 Ntiles x 192 Ksteps)
//   masked softmax over 64 columns (LDS)
//   GEMM2: att[16,768]   = attn[16,64] x SCE[64,768]   (48 Ntiles x 16 Ksteps)
// A layout (16x4 f32): lanes 0-15 -> K={k,k+1}, lanes 16-31 -> K={k+2,k+3}.
// D layout: 8 VGPRs; M = r + 8*(lane>=16), N = lane&15.
// =====================================================================
__global__ __launch_bounds__(32)
void attn_wmma_kernel(const float* __restrict__ q,        // [BS, AT, DIM]
                      const float* __restrict__ sce,      // [BS, CAP, DIM]
                      const int*   __restrict__ cap_len,  // [BS]
                      float*       __restrict__ att_out)  // [BS, AT, DIM]
{
    const int wave  = blockIdx.x;        // BS * (AT/16) waves
    const int b     = wave >> 1;
    const int m0    = (wave & 1) * 16;
    const int lane  = threadIdx.x;
    const int mlane = lane & 15;
    const int khalf = lane >> 4;         // 0 or 1

    const float* __restrict__ Q  = q   + ((size_t)b * AT + m0) * DIM;
    const float* __restrict__ KV = sce + (size_t)b * CAP * DIM;

    // ---------------- GEMM 1: scores = Q * KV^T ----------------
    v8f acc0 = {}, acc1 = {}, acc2 = {}, acc3 = {};
    for (int k = 0; k < DIM; k += 4) {
        const int kk = k + khalf * 2;
        v2f a;
        a.x = Q[mlane * DIM + kk];
        a.y = Q[mlane * DIM + kk + 1];
        v2f b0, b1, b2, b3;
        b0.x = KV[(0 * 16 + mlane) * DIM + kk]; b0.y = KV[(0 * 16 + mlane) * DIM + kk + 1];
        b1.x = KV[(1 * 16 + mlane) * DIM + kk]; b1.y = KV[(1 * 16 + mlane) * DIM + kk + 1];
        b2.x = KV[(2 * 16 + mlane) * DIM + kk]; b2.y = KV[(2 * 16 + mlane) * DIM + kk + 1];
        b3.x = KV[(3 * 16 + mlane) * DIM + kk]; b3.y = KV[(3 * 16 + mlane) * DIM + kk + 1];
        acc0 = __builtin_amdgcn_wmma_f32_16x16x4_f32(false, a, false, b0, (short)0, acc0, false, false);
        acc1 = __builtin_amdgcn_wmma_f32_16x16x4_f32(false, a, false, b1, (short)0, acc1, false, false);
        acc2 = __builtin_amdgcn_wmma_f32_16x16x4_f32(false, a, false, b2, (short)0, acc2, false, false);
        acc3 = __builtin_amdgcn_wmma_f32_16x16x4_f32(false, a, false, b3, (short)0, acc3, false, false);
    }

    // ---------------- masked softmax over the 64 columns ----------------
    __shared__ float s_attn[16][CAP];
    const float scale = 0.03608439182435161f;   // 1/sqrt(768)
    const int clen = cap_len[b];
#pragma unroll
    for (int r = 0; r < 8; ++r) {
        const int m = r + 8 * khalf;
        float s0 = acc0[r] * scale, s1 = acc1[r] * scale,
              s2 = acc2[r] * scale, s3 = acc3[r] * scale;
        if (0 * 16 + mlane >= clen) s0 = -1e9f;
        if (1 * 16 + mlane >= clen) s1 = -1e9f;
        if (2 * 16 + mlane >= clen) s2 = -1e9f;
        if (3 * 16 + mlane >= clen) s3 = -1e9f;
        s_attn[m][0 * 16 + mlane] = s0;
        s_attn[m][1 * 16 + mlane] = s1;
        s_attn[m][2 * 16 + mlane] = s2;
        s_attn[m][3 * 16 + mlane] = s3;
    }
    __syncthreads();

    if (lane < 16) {
        float mx = -INFINITY;
        for (int n = 0; n < CAP; ++n) mx = fmaxf(mx, s_attn[lane][n]);
        float sum = 0.f;
        for (int n = 0; n < CAP; ++n) {
            float e = __expf(s_attn[lane][n] - mx);
            s_attn[lane][n] = e;
            sum += e;
        }
        float inv = 1.0f / sum;
        for (int n = 0; n < CAP; ++n) s_attn[lane][n] *= inv;
    }
    __syncthreads();

    // ---------------- GEMM 2: att = attn * KV ----------------
    float* __restrict__ out = att_out + ((size_t)b * AT + m0) * DIM;
    for (int nt = 0; nt < DIM / 16; ++nt) {       // 48 N-tiles
        const int n0 = nt * 16;
        v8f c = {};
        for (int k = 0; k < CAP; k += 4) {        // 16 K-steps
            const int kk = k + khalf * 2;
            v2f a;
            a.x = s_attn[mlane][kk];
            a.y = s_attn[mlane][kk + 1];
            v2f bf;
            bf.x = KV[(size_t)kk * DIM + n0 + mlane];
            bf.y = KV[(size_t)(kk + 1) * DIM + n0 + mlane];
            c = __builtin_amdgcn_wmma_f32_16x16x4_f32(false, a, false, bf, (short)0, c, false, false);
        }
#pragma unroll
        for (int r = 0; r < 8; ++r)
            out[(size_t)(r + 8 * khalf) * DIM + n0 + mlane] = c[r];
    }
}

// =====================================================================
// Kernel 3: per-(b,s) row select: att_out row / END embedding / token
// embedding. Pure bandwidth copy; one 256-thread block per row.
// =====================================================================
__global__ void scatter_kernel(const float* __restrict__ embed,
                               const float* __restrict__ att_out,
                               const int*   __restrict__ input_ids,
                               const int*   __restrict__ origin_len,
                               const int*   __restrict__ target_len,
                               float*       __restrict__ out)
{
    const int rowid = blockIdx.x;       // b*SEQ + s
    const int b = rowid / SEQ;
    const int s = rowid % SEQ;
    const int rel = s - origin_len[b];
    const int tl  = target_len[b];

    const float* __restrict__ src;
    if (rel >= 0 && rel < tl) {
        int rc = rel < 0 ? 0 : (rel > AT - 1 ? AT - 1 : rel);
        src = att_out + ((size_t)b * AT + rc) * DIM;
    } else if (rel == tl) {
        src = embed + (size_t)END_ID * DIM;
    } else {
        src = embed + (size_t)input_ids[rowid] * DIM;
    }
    float* __restrict__ o = out + (size_t)rowid * DIM;
    for (int d = threadIdx.x; d < DIM; d += 256) o[d] = src[d];
}

// =====================================================================
extern "C" void kernel_launch(void* const* d_in, const int* in_sizes, int n_in,
                              void* d_out, int out_size, void* d_ws, size_t ws_size,
                              hipStream_t stream) {
    const float* caption_out      = (const float*)d_in[0]; // [BS,CAP,VOCAB]
    const float* embed_table      = (const float*)d_in[1]; // [VOCAB,DIM]
    const float* inputs_embeds_at = (const float*)d_in[2]; // [BS,AT,DIM]
    const int*   input_ids        = (const int*)  d_in[3]; // [BS,SEQ]
    const int*   origin_len       = (const int*)  d_in[4]; // [BS]
    const int*   target_len       = (const int*)  d_in[5]; // [BS]
    const int*   caption_len      = (const int*)  d_in[6]; // [BS]
    float*       out              = (float*)d_out;         // [BS,SEQ,DIM]

    float* sce = (float*)d_ws;                             // [BS,CAP,DIM]  3.0 MB
    float* att = sce + (size_t)BS * CAP * DIM;             // [BS,AT,DIM]   1.5 MB

    topk_sce_kernel<<<BS * CAP, 256, 0, stream>>>(caption_out, embed_table, sce);
    attn_wmma_kernel<<<BS * (AT / 16), 32, 0, stream>>>(inputs_embeds_at, sce,
                                                        caption_len, att);
    scatter_kernel<<<BS * SEQ, 256, 0, stream>>>(embed_table, att, input_ids,
                                                 origin_len, target_len, out);
}